// DGT_concat_61211873903543
// MI455X (gfx1250) — compile-verified
//
#include <hip/hip_runtime.h>
#include <hip/hip_bf16.h>

// ---------------------------------------------------------------------------
// DGT layer forward for MI455X (gfx1250, wave32).
// Dense GEMMs: V_WMMA_F32_16X16X4_F32, one wave computes a 32x64 tile
// (2 M-tiles x 4 N-tiles = 8 accumulators; A fragments amortized over N,
// B fragments amortized over M -> ~10.7 flops/byte of cache traffic).
// Irregular graph ops use L2 float atomics + ordered-int atomicMax.
// Workspace: ~2.5 GB of d_ws.
// ---------------------------------------------------------------------------

#define NN_ 10000
#define NE_ 160000
#define ND_ 256
#define ED_ 64
#define TD_ 256
#define H_  8
#define EX_ 2
#define HT_ 10
#define DH_ 32
#define HIK 577          // 2*ND + ED + 1

#define CEILDIV(a,b) (((a)+(b)-1)/(b))

typedef __attribute__((ext_vector_type(2))) float v2f;
typedef __attribute__((ext_vector_type(8))) float v8f;

__device__ __forceinline__ v8f wmma4(v2f a, v2f b, v8f c) {
  // D = A(16x4 f32) * B(4x16 f32) + C(16x16 f32)
  return __builtin_amdgcn_wmma_f32_16x16x4_f32(
      /*neg_a=*/false, a, /*neg_b=*/false, b,
      /*c_mod=*/(short)0, c, /*reuse_a=*/false, /*reuse_b=*/false);
}

__device__ __forceinline__ float dev_silu(float x) { return x / (1.f + __expf(-x)); }

// Ordered-int mapping so unsigned atomicMax == float max.
__device__ __forceinline__ unsigned f2ord(float f) {
  unsigned u = __float_as_uint(f);
  return (u >> 31) ? ~u : (u | 0x80000000u);
}
__device__ __forceinline__ float ord2f(unsigned e) {
  unsigned u = (e >> 31) ? (e & 0x7fffffffu) : ~e;
  return __uint_as_float(u);
}

// ---------------------------------------------------------------------------
// fp32 WMMA GEMM: C[M,N] = A[M,K] @ W[K,N] + bias, row-major.
// One wave computes a 32x64 tile: 8 x (16x16) accumulators. The 4 B fragments
// per K-step are shared by both M sub-tiles; the 2 A fragments are shared by
// all 4 N sub-tiles. M%16==0 and N%64==0 required (true at all call sites);
// K arbitrary (zero-padded tail handles K=577). If M%32==16, the last block
// clamps its second M-tile's loads (redundant compute) and skips its stores.
// ISA layouts (7.12.2): A lanes 0-15 hold M=lane with K={k,k+1} in the two
// VGPRs, lanes 16-31 hold the same rows with K={k+2,k+3}. B symmetric over N.
// C/D: VGPR v -> row (m + 8*half + v), col (n + lane%16).
// ---------------------------------------------------------------------------
template <int ACT>   // 0 = none, 1 = SiLU epilogue
__global__ void k_wmma_gemm(const float* __restrict__ A, int lda,
                            const float* __restrict__ W, int ldw,
                            const float* __restrict__ bias,
                            float* __restrict__ C, int ldc, int M, int K) {
  const int m0   = blockIdx.x << 5;     // 32-row M tile
  const int n0   = blockIdx.y << 6;     // 64-wide N strip
  const int lane = threadIdx.x & 31;
  const int half = lane >> 4;
  const int hl   = lane & 15;
  const int kb   = 2 * half;

  const bool has2 = (m0 + 16) < M;      // second M sub-tile valid?
  const int  r0   = m0 + hl;
  const int  r1   = has2 ? (m0 + 16 + hl) : r0;   // clamp keeps loads in-bounds

  const float* Ar0 = A + (size_t)r0 * lda + kb;
  const float* Ar1 = A + (size_t)r1 * lda + kb;
  const float* B0  = W + (size_t)kb * ldw + (n0 + hl);

  v8f acc[8];
#pragma unroll
  for (int t = 0; t < 8; ++t) acc[t] = (v8f){0.f, 0.f, 0.f, 0.f, 0.f, 0.f, 0.f, 0.f};

  const int K4 = K & ~3;
#pragma unroll 2
  for (int k = 0; k < K4; k += 4) {
    v2f a0, a1;
    a0.x = Ar0[k]; a0.y = Ar0[k + 1];
    a1.x = Ar1[k]; a1.y = Ar1[k + 1];
    const float* Bk0 = B0 + (size_t)k * ldw;   // row k+kb
    const float* Bk1 = Bk0 + ldw;              // row k+kb+1
    v2f b[4];
#pragma unroll
    for (int t = 0; t < 4; ++t) { b[t].x = Bk0[t * 16]; b[t].y = Bk1[t * 16]; }
#pragma unroll
    for (int t = 0; t < 4; ++t) acc[t] = wmma4(a0, b[t], acc[t]);
#pragma unroll
    for (int t = 0; t < 4; ++t) acc[4 + t] = wmma4(a1, b[t], acc[4 + t]);
  }
  if (K4 < K) {                                 // zero-padded K tail (K=577)
    const int i0 = K4 + kb, i1 = i0 + 1;
    v2f a0, a1;
    a0.x = (i0 < K) ? A[(size_t)r0 * lda + i0] : 0.f;
    a0.y = (i1 < K) ? A[(size_t)r0 * lda + i1] : 0.f;
    a1.x = (i0 < K) ? A[(size_t)r1 * lda + i0] : 0.f;
    a1.y = (i1 < K) ? A[(size_t)r1 * lda + i1] : 0.f;
#pragma unroll
    for (int t = 0; t < 4; ++t) {
      v2f b;
      b.x = (i0 < K) ? W[(size_t)i0 * ldw + (n0 + t * 16 + hl)] : 0.f;
      b.y = (i1 < K) ? W[(size_t)i1 * ldw + (n0 + t * 16 + hl)] : 0.f;
      acc[t]     = wmma4(a0, b, acc[t]);
      acc[4 + t] = wmma4(a1, b, acc[4 + t]);
    }
  }

  float bb[4];
#pragma unroll
  for (int t = 0; t < 4; ++t) bb[t] = bias ? bias[n0 + t * 16 + hl] : 0.f;

  float* Cp0 = C + (size_t)(m0 + half * 8) * ldc + (n0 + hl);
#pragma unroll
  for (int t = 0; t < 4; ++t) {
#pragma unroll
    for (int v = 0; v < 8; ++v) {
      float x = acc[t][v] + bb[t];
      if (ACT == 1) x = dev_silu(x);
      Cp0[(size_t)v * ldc + t * 16] = x;
    }
  }
  if (has2) {
    float* Cp1 = Cp0 + (size_t)16 * ldc;
#pragma unroll
    for (int t = 0; t < 4; ++t) {
#pragma unroll
      for (int v = 0; v < 8; ++v) {
        float x = acc[4 + t][v] + bb[t];
        if (ACT == 1) x = dev_silu(x);
        Cp1[(size_t)v * ldc + t * 16] = x;
      }
    }
  }
}

// ---------------------------------------------------------------------------
// LayerNorm (no affine) + adaLN modulation, with optional fused gated
// residual input (x = X1 + G*X2) and optional node mask.
// One block per row, blockDim == D, dynamic LDS = D floats.
// Safe in-place (out == X1): reads complete before writes.
// ---------------------------------------------------------------------------
__global__ void k_ln_mod(const float* __restrict__ X1, const float* __restrict__ X2,
                         const float* __restrict__ G,
                         const float* __restrict__ SH, const float* __restrict__ SC,
                         int ldt, const float* __restrict__ mask,
                         float* __restrict__ out, int D) {
  extern __shared__ float sred[];
  const int row = blockIdx.x;
  const int c   = threadIdx.x;
  const size_t ro = (size_t)row * D + c;
  float x = X1[ro];
  if (X2) x += G[(size_t)row * ldt + c] * X2[ro];

  sred[c] = x;
  __syncthreads();
  for (int st = D >> 1; st > 0; st >>= 1) {
    if (c < st) sred[c] += sred[c + st];
    __syncthreads();
  }
  const float mean = sred[0] / (float)D;
  __syncthreads();
  const float d = x - mean;
  sred[c] = d * d;
  __syncthreads();
  for (int st = D >> 1; st > 0; st >>= 1) {
    if (c < st) sred[c] += sred[c + st];
    __syncthreads();
  }
  const float var = sred[0] / (float)D;

  float y = (x - mean) * rsqrtf(var + 1e-6f);
  const size_t to = (size_t)row * ldt + c;
  y = y * (1.f + SC[to]) + SH[to];
  if (mask) y *= mask[row];
  out[ro] = y;
}

// ---------------------------------------------------------------------------
// Elementwise / graph kernels
// ---------------------------------------------------------------------------
__global__ void k_silu_arr(const float* __restrict__ x, float* __restrict__ y, int n) {
  int i = blockIdx.x * blockDim.x + threadIdx.x;
  if (i < n) y[i] = dev_silu(x[i]);
}

__global__ void k_dist(const float* __restrict__ pos, const int* __restrict__ row,
                       const int* __restrict__ col, float* __restrict__ dist) {
  int e = blockIdx.x * blockDim.x + threadIdx.x;
  if (e >= NE_) return;
  int r = row[e], c = col[e];
  float dx = pos[r * 3 + 0] - pos[c * 3 + 0];
  float dy = pos[r * 3 + 1] - pos[c * 3 + 1];
  float dz = pos[r * 3 + 2] - pos[c * 3 + 2];
  dist[e] = dx * dx + dy * dy + dz * dz;
}

// ea = concat(dist, edge_attr) @ edge_emb_w + b   (K=65 -> plain VALU)
__global__ void k_edge_emb(const float* __restrict__ eattr, const float* __restrict__ dist,
                           const float* __restrict__ w, const float* __restrict__ b,
                           float* __restrict__ ea) {
  int i = blockIdx.x * blockDim.x + threadIdx.x;
  if (i >= NE_ * ED_) return;
  int e = i >> 6, c = i & 63;
  float acc = b[c] + dist[e] * w[c];
  const float* ar = eattr + (size_t)e * ED_;
#pragma unroll 8
  for (int k = 0; k < ED_; ++k) acc += ar[k] * w[(k + 1) * ED_ + c];
  ea[i] = acc;
}

// logits[e,h] (h<8: scaled q.(k+ek); h>=8: extra heads) + atomicMax per (row,h)
__global__ void k_logits(const float* __restrict__ q, const float* __restrict__ kk,
                         const float* __restrict__ ek, const float* __restrict__ extra,
                         const int* __restrict__ row, const int* __restrict__ col,
                         float* __restrict__ logits, unsigned* __restrict__ nmax) {
  int i = blockIdx.x * blockDim.x + threadIdx.x;
  if (i >= NE_ * HT_) return;
  int e = i / HT_, hh = i - e * HT_;
  float l;
  if (hh < H_) {
    int r = row[e], c = col[e];
    const float* qp = q + (size_t)r * ND_ + hh * DH_;
    const float* kp = kk + (size_t)c * ND_ + hh * DH_;
    const float* ep = ek + (size_t)e * ND_ + hh * DH_;
    float s = 0.f;
#pragma unroll
    for (int d = 0; d < DH_; ++d) s += qp[d] * (kp[d] + ep[d]);
    l = s * 0.17677669529663687f;  // 1/sqrt(32)
  } else {
    l = extra[(size_t)e * EX_ + (hh - H_)];
  }
  logits[i] = l;
  atomicMax(&nmax[(size_t)row[e] * HT_ + hh], f2ord(l));
}

// p = exp(l - max); denom[row,h] += p; agg[row,h,:] += p*(v[col,h,:]+ev[e,h,:])
__global__ void k_pass2(const float* __restrict__ logits, const unsigned* __restrict__ nmax,
                        const float* __restrict__ v, const float* __restrict__ ev,
                        const int* __restrict__ row, const int* __restrict__ col,
                        float* __restrict__ denom, float* __restrict__ agg) {
  int i = blockIdx.x * blockDim.x + threadIdx.x;
  if (i >= NE_ * HT_) return;
  int e = i / HT_, hh = i - e * HT_;
  int r = row[e], c = col[e];
  float m = ord2f(nmax[(size_t)r * HT_ + hh]);
  float p = __expf(logits[i] - m);
  atomicAdd(&denom[(size_t)r * HT_ + hh], p);
  const float* vp = v + (size_t)c * (HT_ * DH_) + hh * DH_;
  const float* ep = ev + (size_t)e * (HT_ * DH_) + hh * DH_;
  float* ap = agg + (size_t)r * (HT_ * DH_) + hh * DH_;
#pragma unroll
  for (int d = 0; d < DH_; ++d) atomicAdd(&ap[d], p * (vp[d] + ep[d]));
}

__global__ void k_norm(float* __restrict__ agg, const float* __restrict__ denom) {
  int i = blockIdx.x * blockDim.x + threadIdx.x;
  if (i >= NN_ * HT_ * DH_) return;
  int n = i / (HT_ * DH_), cc = i - n * (HT_ * DH_);
  agg[i] /= (denom[(size_t)n * HT_ + cc / DH_] + 1e-16f);
}

__global__ void k_sumrc(const float* __restrict__ hatt, const int* __restrict__ row,
                        const int* __restrict__ col, float* __restrict__ out) {
  int i = blockIdx.x * blockDim.x + threadIdx.x;
  if (i >= NE_ * ND_) return;
  int e = i >> 8, c = i & 255;
  out[i] = hatt[(size_t)row[e] * ND_ + c] + hatt[(size_t)col[e] * ND_ + c];
}

__global__ void k_node_out(const float* __restrict__ hmod, const float* __restrict__ nt,
                           const float* __restrict__ ffn, const float* __restrict__ mask,
                           float* __restrict__ out) {
  int i = blockIdx.x * blockDim.x + threadIdx.x;
  if (i >= NN_ * ND_) return;
  int n = i >> 8, c = i & 255;
  out[i] = (hmod[i] + nt[(size_t)n * (6 * ND_) + 5 * ND_ + c] * ffn[i]) * mask[n];
}

__global__ void k_edge_out(const float* __restrict__ em2, const float* __restrict__ et,
                           const float* __restrict__ ffe, float* __restrict__ out) {
  int i = blockIdx.x * blockDim.x + threadIdx.x;
  if (i >= NE_ * ED_) return;
  int e = i >> 6, c = i & 63;
  out[i] = em2[i] + et[(size_t)e * (6 * ED_) + 5 * ED_ + c] * ffe[i];
}

// hi = concat(h_out[row], h_out[col], h_edge_out, dist)   [NE, 577]
__global__ void k_build_hi(const float* __restrict__ outh, const float* __restrict__ oute,
                           const float* __restrict__ dist, const int* __restrict__ row,
                           const int* __restrict__ col, float* __restrict__ hi) {
  int i = blockIdx.x * blockDim.x + threadIdx.x;
  if (i >= NE_ * HIK) return;
  int e = i / HIK, c = i - e * HIK;
  float v;
  if (c < ND_)            v = outh[(size_t)row[e] * ND_ + c];
  else if (c < 2 * ND_)   v = outh[(size_t)col[e] * ND_ + (c - ND_)];
  else if (c < HIK - 1)   v = oute[(size_t)e * ED_ + (c - 2 * ND_)];
  else                    v = dist[e];
  hi[i] = v;
}

// inv = mean(tanh(c1 @ eq_c2_w) * [1, eh0, eh1])
__global__ void k_eqc2(const float* __restrict__ c1, const float* __restrict__ w2,
                       const float* __restrict__ extra, float* __restrict__ invs) {
  int e = blockIdx.x * blockDim.x + threadIdx.x;
  if (e >= NE_) return;
  const float* cp = c1 + (size_t)e * ND_;
  float t0 = 0.f, t1 = 0.f, t2 = 0.f;
#pragma unroll 4
  for (int k = 0; k < ND_; ++k) {
    float x = cp[k];
    t0 += x * w2[k * 3 + 0];
    t1 += x * w2[k * 3 + 1];
    t2 += x * w2[k * 3 + 2];
  }
  t0 = tanhf(t0); t1 = tanhf(t1); t2 = tanhf(t2);
  invs[e] = (t0 + t1 * extra[(size_t)e * EX_ + 0] + t2 * extra[(size_t)e * EX_ + 1]) * (1.f / 3.f);
}

__global__ void k_pos_copy(const float* __restrict__ pos, float* __restrict__ outp) {
  int i = blockIdx.x * blockDim.x + threadIdx.x;
  if (i < NN_ * 3) outp[i] = pos[i];
}

__global__ void k_pos_scatter(const float* __restrict__ pos, const int* __restrict__ row,
                              const int* __restrict__ col, const float* __restrict__ invs,
                              const float* __restrict__ scale, float* __restrict__ outp) {
  int e = blockIdx.x * blockDim.x + threadIdx.x;
  if (e >= NE_) return;
  int r = row[e], c = col[e];
  float dx = pos[r * 3 + 0] - pos[c * 3 + 0];
  float dy = pos[r * 3 + 1] - pos[c * 3 + 1];
  float dz = pos[r * 3 + 2] - pos[c * 3 + 2];
  float nrm = sqrtf(dx * dx + dy * dy + dz * dz);
  float s = scale[0] * invs[e] / fmaxf(nrm, 1e-8f);
  atomicAdd(&outp[r * 3 + 0], dx * s);
  atomicAdd(&outp[r * 3 + 1], dy * s);
  atomicAdd(&outp[r * 3 + 2], dz * s);
}

// ---------------------------------------------------------------------------
// Host side
// ---------------------------------------------------------------------------
static void gemm(hipStream_t s, const float* A, int lda, const float* W, int ldw,
                 const float* bias, float* C, int ldc, int M, int N, int K, bool act) {
  dim3 g(CEILDIV(M, 32), N / 64);   // N%64==0 at every call site
  if (act) k_wmma_gemm<1><<<g, dim3(32), 0, s>>>(A, lda, W, ldw, bias, C, ldc, M, K);
  else     k_wmma_gemm<0><<<g, dim3(32), 0, s>>>(A, lda, W, ldw, bias, C, ldc, M, K);
}

extern "C" void kernel_launch(void* const* d_in, const int* in_sizes, int n_in,
                              void* d_out, int out_size, void* d_ws, size_t ws_size,
                              hipStream_t stream) {
  (void)in_sizes; (void)n_in; (void)out_size; (void)ws_size;

  const float* pos   = (const float*)d_in[0];
  const float* h     = (const float*)d_in[1];
  const float* eattr = (const float*)d_in[2];
  const int*   eidx  = (const int*)d_in[3];
  const float* nmask = (const float*)d_in[4];
  const float* extra = (const float*)d_in[5];
  const float* nte   = (const float*)d_in[6];
  const float* ete   = (const float*)d_in[7];
  const int* erow = eidx;
  const int* ecol = eidx + NE_;

  const float* p_edge_emb_w = (const float*)d_in[8];
  const float* p_edge_emb_b = (const float*)d_in[9];
  const float* p_ntime_w    = (const float*)d_in[10];
  const float* p_ntime_b    = (const float*)d_in[11];
  const float* p_etime_w    = (const float*)d_in[12];
  const float* p_etime_b    = (const float*)d_in[13];
  const float* p_wq  = (const float*)d_in[14];
  const float* p_bq  = (const float*)d_in[15];
  const float* p_wk  = (const float*)d_in[16];
  const float* p_bk  = (const float*)d_in[17];
  const float* p_wek = (const float*)d_in[18];
  const float* p_bek = (const float*)d_in[19];
  const float* p_wv  = (const float*)d_in[20];
  const float* p_bv  = (const float*)d_in[21];
  const float* p_wev = (const float*)d_in[22];
  const float* p_bev = (const float*)d_in[23];
  const float* p_wo  = (const float*)d_in[24];
  const float* p_bo  = (const float*)d_in[25];
  const float* p_n2e_w = (const float*)d_in[26];
  const float* p_n2e_b = (const float*)d_in[27];
  const float* p_ff1_w = (const float*)d_in[28];
  const float* p_ff1_b = (const float*)d_in[29];
  const float* p_ff2_w = (const float*)d_in[30];
  const float* p_ff2_b = (const float*)d_in[31];
  const float* p_ff3_w = (const float*)d_in[32];
  const float* p_ff3_b = (const float*)d_in[33];
  const float* p_ff4_w = (const float*)d_in[34];
  const float* p_ff4_b = (const float*)d_in[35];
  const float* p_eqt_w  = (const float*)d_in[36];
  const float* p_eqt_b  = (const float*)d_in[37];
  const float* p_eqin_w = (const float*)d_in[38];
  const float* p_eqin_b = (const float*)d_in[39];
  const float* p_eqc1_w = (const float*)d_in[40];
  const float* p_eqc1_b = (const float*)d_in[41];
  const float* p_eqc2_w = (const float*)d_in[42];
  const float* p_coors  = (const float*)d_in[43];

  float* out_h = (float*)d_out;
  float* out_e = out_h + (size_t)NN_ * ND_;
  float* out_p = out_e + (size_t)NE_ * ED_;

  // Workspace bump allocator (floats), 256-element aligned.
  float* wsf = (float*)d_ws;
  size_t off = 0;
  auto alloc = [&](size_t n) -> float* {
    float* p = wsf + off;
    off += (n + 255) & ~(size_t)255;
    return p;
  };

  float*    dist   = alloc(NE_);
  float*    ea     = alloc((size_t)NE_ * ED_);
  float*    stn    = alloc((size_t)NN_ * TD_);           // silu(node_time_emb)
  float*    ste    = alloc((size_t)NE_ * TD_);           // silu(edge_time_emb)
  float*    nt     = alloc((size_t)NN_ * 6 * ND_);
  float*    et     = alloc((size_t)NE_ * 6 * ED_);
  float*    eqt    = alloc((size_t)NE_ * 2 * ND_);
  float*    hm     = alloc((size_t)NN_ * ND_);
  float*    em     = alloc((size_t)NE_ * ED_);
  float*    q      = alloc((size_t)NN_ * ND_);
  float*    kk     = alloc((size_t)NN_ * ND_);
  float*    v      = alloc((size_t)NN_ * HT_ * DH_);
  float*    ek     = alloc((size_t)NE_ * ND_);
  float*    ev     = alloc((size_t)NE_ * HT_ * DH_);
  float*    logits = alloc((size_t)NE_ * HT_);
  unsigned* nmax   = (unsigned*)alloc((size_t)NN_ * HT_);
  float*    denom  = alloc((size_t)NN_ * HT_);
  float*    agg    = alloc((size_t)NN_ * HT_ * DH_);
  float*    hatt   = alloc((size_t)NN_ * ND_);
  float*    sumrc  = alloc((size_t)NE_ * ND_);
  float*    heraw  = alloc((size_t)NE_ * ED_);
  float*    hmod   = alloc((size_t)NN_ * ND_);
  float*    ffnh   = alloc((size_t)NN_ * 2 * ND_);
  float*    ffn    = alloc((size_t)NN_ * ND_);
  float*    em2    = alloc((size_t)NE_ * ED_);
  float*    ffeh   = alloc((size_t)NE_ * 2 * ED_);
  float*    ffe    = alloc((size_t)NE_ * ED_);
  float*    hi     = alloc((size_t)NE_ * HIK);
  float*    inv1   = alloc((size_t)NE_ * ND_);
  float*    c1buf  = alloc((size_t)NE_ * ND_);
  float*    invs   = alloc(NE_);

  // 1) SiLU of time embeddings (materialized so the big GEMMs reload cheap data)
  k_silu_arr<<<CEILDIV(NN_ * TD_, 256), 256, 0, stream>>>(nte, stn, NN_ * TD_);
  k_silu_arr<<<CEILDIV(NE_ * TD_, 256), 256, 0, stream>>>(ete, ste, NE_ * TD_);

  // 2) adaLN time-conditioning GEMMs (WMMA)
  gemm(stream, stn, TD_, p_ntime_w, 6 * ND_, p_ntime_b, nt, 6 * ND_, NN_, 6 * ND_, TD_, false);
  gemm(stream, ste, TD_, p_etime_w, 6 * ED_, p_etime_b, et, 6 * ED_, NE_, 6 * ED_, TD_, false);
  gemm(stream, ste, TD_, p_eqt_w, 2 * ND_, p_eqt_b, eqt, 2 * ND_, NE_, 2 * ND_, TD_, false);

  // 3) dist + edge embedding
  k_dist<<<CEILDIV(NE_, 256), 256, 0, stream>>>(pos, erow, ecol, dist);
  k_edge_emb<<<CEILDIV(NE_ * ED_, 256), 256, 0, stream>>>(eattr, dist, p_edge_emb_w, p_edge_emb_b, ea);

  // 4) LN + modulation of node/edge features
  k_ln_mod<<<NN_, ND_, ND_ * sizeof(float), stream>>>(h, nullptr, nullptr,
      nt + 0 * ND_, nt + 1 * ND_, 6 * ND_, nullptr, hm, ND_);
  k_ln_mod<<<NE_, ED_, ED_ * sizeof(float), stream>>>(ea, nullptr, nullptr,
      et + 0 * ED_, et + 1 * ED_, 6 * ED_, nullptr, em, ED_);

  // 5) q/k/v and edge-conditioned k/v projections (WMMA)
  gemm(stream, hm, ND_, p_wq, ND_, p_bq, q, ND_, NN_, ND_, ND_, false);
  gemm(stream, hm, ND_, p_wk, ND_, p_bk, kk, ND_, NN_, ND_, ND_, false);
  gemm(stream, hm, ND_, p_wv, HT_ * DH_, p_bv, v, HT_ * DH_, NN_, HT_ * DH_, ND_, false);
  gemm(stream, em, ED_, p_wek, ND_, p_bek, ek, ND_, NE_, ND_, ED_, false);
  gemm(stream, em, ED_, p_wev, HT_ * DH_, p_bev, ev, HT_ * DH_, NE_, HT_ * DH_, ED_, false);

  // 6) segment softmax + aggregation (fused normalization)
  hipMemsetAsync(nmax, 0, (size_t)NN_ * HT_ * sizeof(unsigned), stream);  // ordered-int -inf
  hipMemsetAsync(denom, 0, (size_t)NN_ * HT_ * sizeof(float), stream);
  hipMemsetAsync(agg, 0, (size_t)NN_ * HT_ * DH_ * sizeof(float), stream);
  k_logits<<<CEILDIV(NE_ * HT_, 256), 256, 0, stream>>>(q, kk, ek, extra, erow, ecol, logits, nmax);
  k_pass2<<<CEILDIV(NE_ * HT_, 256), 256, 0, stream>>>(logits, nmax, v, ev, erow, ecol, denom, agg);
  k_norm<<<CEILDIV(NN_ * HT_ * DH_, 256), 256, 0, stream>>>(agg, denom);
  gemm(stream, agg, HT_ * DH_, p_wo, ND_, p_bo, hatt, ND_, NN_, ND_, HT_ * DH_, false);

  // 7) edge update from attended nodes
  k_sumrc<<<CEILDIV(NE_ * ND_, 256), 256, 0, stream>>>(hatt, erow, ecol, sumrc);
  gemm(stream, sumrc, ND_, p_n2e_w, ED_, p_n2e_b, heraw, ED_, NE_, ED_, ND_, false);

  // 8) node residual + gated FFN -> h_out
  k_ln_mod<<<NN_, ND_, ND_ * sizeof(float), stream>>>(h, hatt, nt + 2 * ND_,
      nt + 3 * ND_, nt + 4 * ND_, 6 * ND_, nmask, hmod, ND_);
  gemm(stream, hmod, ND_, p_ff1_w, 2 * ND_, p_ff1_b, ffnh, 2 * ND_, NN_, 2 * ND_, ND_, true);
  gemm(stream, ffnh, 2 * ND_, p_ff2_w, ND_, p_ff2_b, ffn, ND_, NN_, ND_, 2 * ND_, false);
  k_node_out<<<CEILDIV(NN_ * ND_, 256), 256, 0, stream>>>(hmod, nt, ffn, nmask, out_h);

  // 9) edge residual + gated FFN -> h_edge_out
  k_ln_mod<<<NE_, ED_, ED_ * sizeof(float), stream>>>(eattr, heraw, et + 2 * ED_,
      et + 3 * ED_, et + 4 * ED_, 6 * ED_, nullptr, em2, ED_);
  gemm(stream, em2, ED_, p_ff3_w, 2 * ED_, p_ff3_b, ffeh, 2 * ED_, NE_, 2 * ED_, ED_, true);
  gemm(stream, ffeh, 2 * ED_, p_ff4_w, ED_, p_ff4_b, ffe, ED_, NE_, ED_, 2 * ED_, false);
  k_edge_out<<<CEILDIV(NE_ * ED_, 256), 256, 0, stream>>>(em2, et, ffe, out_e);

  // 10) equivariant coordinate update -> pos_out
  k_build_hi<<<CEILDIV(NE_ * HIK, 256), 256, 0, stream>>>(out_h, out_e, dist, erow, ecol, hi);
  gemm(stream, hi, HIK, p_eqin_w, ND_, p_eqin_b, inv1, ND_, NE_, ND_, HIK, false);
  k_ln_mod<<<NE_, ND_, ND_ * sizeof(float), stream>>>(inv1, nullptr, nullptr,
      eqt + 0 * ND_, eqt + 1 * ND_, 2 * ND_, nullptr, inv1, ND_);   // in-place LN+mod
  gemm(stream, inv1, ND_, p_eqc1_w, ND_, p_eqc1_b, c1buf, ND_, NE_, ND_, ND_, true);
  k_eqc2<<<CEILDIV(NE_, 256), 256, 0, stream>>>(c1buf, p_eqc2_w, extra, invs);
  k_pos_copy<<<CEILDIV(NN_ * 3, 256), 256, 0, stream>>>(pos, out_p);
  k_pos_scatter<<<CEILDIV(NE_, 256), 256, 0, stream>>>(pos, erow, ecol, invs, p_coors, out_p);
}